// BiDiBoundaryPredictor_65884798320954
// MI455X (gfx1250) — compile-verified
//
#include <hip/hip_runtime.h>

typedef __attribute__((ext_vector_type(2))) float v2f;
typedef __attribute__((ext_vector_type(8))) float v8f;

#define BSZ   16
#define LEN   8192
#define DIM   512
#define NEGV  (-10000.0f)
#define CHUNK 128
#define CPB   (LEN / CHUNK)        // 64 chunks per batch row
#define NCHUNK (BSZ * CPB)         // 1024 chunks total
#define NROWS ((long)BSZ * LEN)    // 131072 timesteps

// ---------------------------------------------------------------------------
// Kernel 1: logits = x @ W + b via V_WMMA_F32_16X16X4_F32 (exact fp32 path).
// One wave32 per 16-timestep tile. A: 16x4 f32 (2 VGPR), B: 4x16 f32 (2 VGPR),
// C: 16x16 f32 (8 VGPR). Only output columns 0,1 are meaningful; B cols >=2
// are zero. W is staged transposed in LDS so the B fragment is one b64 read.
// ---------------------------------------------------------------------------
__global__ __launch_bounds__(256) void policy_gemv_wmma(
    const float* __restrict__ x, const float* __restrict__ W,
    const float* __restrict__ bias, float* __restrict__ logits)
{
    __shared__ float Wt[2 * DIM];                 // Wt[n*DIM + k] = W[k*2 + n]
    const int tid = threadIdx.x;
    for (int i = tid; i < 2 * DIM; i += 256) {
        int k = i >> 1, n = i & 1;
        Wt[n * DIM + k] = W[i];
    }
    __syncthreads();

    const int wave = tid >> 5;
    const int lane = tid & 31;
    const int hf   = lane >> 4;                   // K half: {0,1} vs {2,3}
    const int m    = lane & 15;                   // A row / B-C column

    const long tile = (long)blockIdx.x * 8 + wave;   // 8192 tiles of 16 rows
    const long row0 = tile * 16;

    const float* arow  = x + (row0 + m) * (long)DIM + 2 * hf;
    const bool   bval  = (m < 2);
    const float* bbase = &Wt[(bval ? m : 0) * DIM + 2 * hf];

    v8f acc = {};
    #pragma unroll 4
    for (int kk = 0; kk < DIM; kk += 4) {
        __builtin_prefetch(arow + kk + 128, 0, 0);    // speculative, RT
        float2 av = *(const float2*)(arow + kk);      // global_load_b64
        float2 bv = *(const float2*)(bbase + kk);     // ds_load_b64
        v2f a, b;
        a[0] = av.x;               a[1] = av.y;
        b[0] = bval ? bv.x : 0.f;  b[1] = bval ? bv.y : 0.f;
        // D = A(16x4) * B(4x16) + C ; 8 args: neg_a, A, neg_b, B, c_mod, C, reuse_a, reuse_b
        acc = __builtin_amdgcn_wmma_f32_16x16x4_f32(
                  false, a, false, b, (short)0, acc, false, false);
    }

    if (bval) {                                   // columns 0,1 only
        const float bb = bias[m];
        #pragma unroll
        for (int r = 0; r < 8; ++r) {
            long row = row0 + r + 8 * hf;         // C layout: VGPR r -> M = r + 8*half
            logits[row * 2 + m] = acc[r] + bb;
        }
    }
}

// ---------------------------------------------------------------------------
// FSM scan: flag in {0..3}. Per step: s==0 -> (p?3:0) ; s>0 -> (q?3:s-1)
// where p = l1>l0, q = (l1+NEG)>l0. A map {0..3}->{0..3} packs into 8 bits.
// ---------------------------------------------------------------------------
__device__ __forceinline__ unsigned map_apply(unsigned mp, unsigned s) {
    return (mp >> (2u * s)) & 3u;
}

__global__ void chunk_maps_kernel(const float* __restrict__ logits,
                                  unsigned* __restrict__ maps)
{
    int c = blockIdx.x * blockDim.x + threadIdx.x;
    if (c >= NCHUNK) return;
    long t0 = (long)c * CHUNK;                    // chunks never cross batches
    unsigned r = 0xE4u;                           // identity map 3,2,1,0
    for (int i = 0; i < CHUNK; ++i) {
        float l0 = logits[(t0 + i) * 2 + 0];
        float l1 = logits[(t0 + i) * 2 + 1];
        unsigned p = (l1 > l0) ? 1u : 0u;
        unsigned q = ((l1 + NEGV) > l0) ? 1u : 0u;
        unsigned step = (p ? 3u : 0u)
                      | ((q ? 3u : 0u) << 2)
                      | ((q ? 3u : 1u) << 4)
                      | ((q ? 3u : 2u) << 6);
        unsigned nr = 0;
        #pragma unroll
        for (unsigned s = 0; s < 4; ++s)
            nr |= map_apply(step, map_apply(r, s)) << (2u * s);
        r = nr;
    }
    maps[c] = r;
}

__global__ void scan_maps_kernel(const unsigned* __restrict__ maps,
                                 unsigned* __restrict__ inflags)
{
    int b = threadIdx.x;
    if (b >= BSZ) return;
    unsigned mloc[CPB];
    #pragma unroll
    for (int c = 0; c < CPB; ++c) mloc[c] = maps[b * CPB + c];
    unsigned flag = 0;
    #pragma unroll
    for (int c = 0; c < CPB; ++c) {
        inflags[b * CPB + c] = flag;
        flag = map_apply(mloc[c], flag);
    }
}

__global__ void replay_kernel(const float* __restrict__ logits,
                              const unsigned* __restrict__ inflags,
                              float* __restrict__ out)
{
    int c = blockIdx.x * blockDim.x + threadIdx.x;
    if (c >= NCHUNK) return;
    long t0 = (long)c * CHUNK;
    unsigned s = inflags[c];
    for (int i = 0; i < CHUNK; ++i) {
        long t = t0 + i;
        float l0  = logits[t * 2 + 0];
        float l1  = logits[t * 2 + 1];
        float l1m = (s > 0) ? (l1 + NEGV) : l1;
        bool  pred = l1m > l0;                         // argmax, ties -> 0
        // log_softmax(z)/gather: z = masked/TAU; logp@argmax = -log1p(exp(-|z1-z0|))
        float diff = fabsf((l1m - l0) * 0.5f);
        float g = -log1pf(expf(-diff));
        out[t]         = pred ? 1.0f : 0.0f;           // boundaries [16,8192]
        out[NROWS + t] = g;                            // gathered   [16,8192,1]
        s = pred ? 3u : (s > 0 ? s - 1u : 0u);
    }
}

// ---------------------------------------------------------------------------
extern "C" void kernel_launch(void* const* d_in, const int* in_sizes, int n_in,
                              void* d_out, int out_size, void* d_ws, size_t ws_size,
                              hipStream_t stream) {
    const float* x    = (const float*)d_in[0];   // [16, 8192, 512] f32
    // d_in[1]: labels (unused by forward math)
    const float* W    = (const float*)d_in[2];   // [512, 2] f32
    const float* bias = (const float*)d_in[3];   // [2] f32

    float*    logits  = (float*)d_ws;                                   // 1 MB
    unsigned* maps    = (unsigned*)((char*)d_ws + (1 << 20));           // 4 KB
    unsigned* inflags = maps + NCHUNK;                                  // 4 KB
    float*    out     = (float*)d_out;

    policy_gemv_wmma<<<1024, 256, 0, stream>>>(x, W, bias, logits);
    chunk_maps_kernel<<<(NCHUNK + 255) / 256, 256, 0, stream>>>(logits, maps);
    scan_maps_kernel<<<1, 32, 0, stream>>>(maps, inflags);
    replay_kernel<<<(NCHUNK + 255) / 256, 256, 0, stream>>>(logits, inflags, out);
}